// CombtLinearTCQ_6030134083825
// MI455X (gfx1250) — compile-verified
//
#include <hip/hip_runtime.h>

typedef __attribute__((ext_vector_type(2))) float v2f;
typedef __attribute__((ext_vector_type(8))) float v8f;

constexpr int IN_FEATURES  = 4096;
constexpr int OUT_FEATURES = 11008;
constexpr int KPART = 2048;
constexpr int KO    = 128;                 // k-tiles per partition
constexpr int NW    = 32;                  // 16-bit words per block (one per lane)
constexpr int BATCH = 4;
constexpr int WORDS_PER_PART = KO * NW;    // 4096 words = 16 KB per trellis per mo

#define AS1 __attribute__((address_space(1)))
#define AS3 __attribute__((address_space(3)))

__global__ __launch_bounds__(256, 2)
void tcq_fused_decode_gemm(const float* __restrict__ inp,
                           const int*   __restrict__ trellis1,
                           const int*   __restrict__ trellis2,
                           const float* __restrict__ tlut,
                           float*       __restrict__ out)
{
    __shared__ int   lds_tr[2 * WORDS_PER_PART];  // 32 KB: this WG's full trellis slice
    __shared__ float lds_tlut[512 * 2];           // 4 KB LUT copy
    __shared__ float lds_tile[8][256];            // per-wave 16x16 decoded tile
    __shared__ float lds_acc[BATCH][16];          // cross-wave output reduction

    const int tid  = threadIdx.x;
    const int wave = tid >> 5;
    const int lane = tid & 31;
    const int mo   = blockIdx.x;                  // 16-row output tile index

    // ---- bulk-stage the trellis slice (contiguous 16 KB per partition) ----
    const long long trbase = (long long)mo * WORDS_PER_PART;
#if __has_builtin(__builtin_amdgcn_global_load_async_to_lds_b32)
    {
        AS1 int* g1 = (AS1 int*)(trellis1 + trbase);   // builtin wants non-const
        AS1 int* g2 = (AS1 int*)(trellis2 + trbase);
        AS3 int* l  = (AS3 int*)lds_tr;
        for (int i = tid; i < WORDS_PER_PART; i += 256) {
            __builtin_amdgcn_global_load_async_to_lds_b32(g1 + i, l + i, 0, 0);
            __builtin_amdgcn_global_load_async_to_lds_b32(g2 + i, l + WORDS_PER_PART + i, 0, 0);
        }
#if __has_builtin(__builtin_amdgcn_s_wait_asynccnt)
        __builtin_amdgcn_s_wait_asynccnt(0);
#else
        asm volatile("s_wait_asynccnt 0x0" ::: "memory");
#endif
    }
#else
    {
        const int4* g1 = (const int4*)(trellis1 + trbase);
        const int4* g2 = (const int4*)(trellis2 + trbase);
        int4* l = (int4*)lds_tr;
        for (int i = tid; i < WORDS_PER_PART / 4; i += 256) {   // b128 streams
            l[i]                      = g1[i];
            l[WORDS_PER_PART / 4 + i] = g2[i];
        }
    }
#endif

    for (int i = tid; i < 1024; i += 256) lds_tlut[i] = tlut[i];
    if (tid < 64) lds_acc[tid >> 4][tid & 15] = 0.0f;
    __syncthreads();

    v8f c = {};                                   // fp32 16x16 accumulator (8 VGPRs)

    const int   n    = lane & 15;                 // B column = output row within tile
    const int   hi   = lane >> 4;                 // lane half selects K stripe
    const int   m    = lane & 15;                 // A row = batch (padded 4 -> 16)
    const int   mi   = (m < BATCH) ? m : 0;
    const float asel = (m < BATCH) ? 1.0f : 0.0f;

    for (int t = wave; t < 2 * KO; t += 8) {
        // ---- parallel trellis decode from LDS: one 16-bit word per lane ----
        // idx(s) = ((c[s-2]&1)<<8) | (c[s-1]<<4) | c[s]  (sliding window, no scan)
        const int* twp = &lds_tr[t * NW];
        unsigned w16  = ((unsigned)twp[lane]) & 0xFFFFu;
        unsigned prev = (unsigned)twp[lane ? lane - 1 : 0];   // carry word i-1
        if (lane == 0) prev = 0u;                              // initial state is 0

        unsigned cm2 = (prev >> 4) & 0xFu;        // chunk s-2 carry-in
        unsigned cm1 = prev & 0xFu;               // chunk s-1 carry-in
        float* tilep = &lds_tile[wave][0];
        #pragma unroll
        for (int j = 0; j < 4; ++j) {
            unsigned ch  = (w16 >> (12 - 4 * j)) & 0xFu;       // big-endian chunks
            unsigned idx = ((cm2 & 1u) << 8) | (cm1 << 4) | ch;
            v2f wv = *(const v2f*)&lds_tlut[idx * 2];          // ds_load_b64 gather
            *(v2f*)&tilep[8 * lane + 2 * j] = wv;              // step-major layout
            cm2 = cm1; cm1 = ch;
        }

        // wave-local LDS visibility (only this wave touches lds_tile[wave])
        asm volatile("s_wait_dscnt 0" ::: "memory");
        __builtin_amdgcn_wave_barrier();

        // ---- 4 x V_WMMA_F32_16X16X4_F32 over this 16-wide K slab ----------
        const int part  = t >> 7;
        const int ko    = t & (KO - 1);
        const int gbase = part * KPART + ko * 16;
        #pragma unroll
        for (int q = 0; q < 4; ++q) {
            const int k0 = 4 * q + 2 * hi;        // VGPR pair covers K = k0, k0+1
            // B[k][n] = Wtile[n*16 + k]  (W^T so D columns are output rows)
            v2f b = *(const v2f*)&tilep[n * 16 + k0];
            // A[m][k] = x[batch, gbase + k]; padded batch rows zeroed
            v2f a = *(const v2f*)&inp[(long long)mi * IN_FEATURES + gbase + k0];
            a.x *= asel; a.y *= asel;
            c = __builtin_amdgcn_wmma_f32_16x16x4_f32(false, a, false, b,
                                                      (short)0, c, false, false);
        }
        __builtin_amdgcn_wave_barrier();
    }

    // ---- reduce 8 waves: batches 0..3 live in C VGPRs 0..3, lanes 0-15 ----
    if (lane < 16) {
        #pragma unroll
        for (int b = 0; b < BATCH; ++b)
            atomicAdd(&lds_acc[b][lane], c[b]);   // ds_add_f32
    }
    __syncthreads();

    if (tid < 64) {
        const int b = tid >> 4, nn = tid & 15;
        out[(long long)b * OUT_FEATURES + mo * 16 + nn] = lds_acc[b][nn];
    }
}

extern "C" void kernel_launch(void* const* d_in, const int* in_sizes, int n_in,
                              void* d_out, int out_size, void* d_ws, size_t ws_size,
                              hipStream_t stream) {
    const float* inp  = (const float*)d_in[0];
    const int*   t1   = (const int*)d_in[1];
    const int*   t2   = (const int*)d_in[2];
    const float* tlut = (const float*)d_in[3];
    float*       out  = (float*)d_out;

    dim3 grid(OUT_FEATURES / 16);   // 688 workgroups, one per 16-row output tile
    tcq_fused_decode_gemm<<<grid, 256, 0, stream>>>(inp, t1, t2, tlut, out);
}